// WassersteinLoss_47545287966761
// MI455X (gfx1250) — compile-verified
//
#include <hip/hip_runtime.h>

// ---------------------------------------------------------------------------
// Sinkhorn Wasserstein loss, MI455X (gfx1250, wave32).
// B=8, S=2048, P=1024, D=2, EPS=0.01, 50 iterations.
//
// ||x-y||^2 = x0*(-2y0) + x1*(-2y1) + (x0^2+x1^2)*1 + 1*(y0^2+y1^2) is a K=4
// dot product -> one V_WMMA_F32_16X16X4_F32 produces a 16x16 squared-distance
// tile in registers; the 16.7M-element cost matrix is never materialized.
// Online logsumexp (max,sum) state lives in registers; cross-lane reduction
// only at sweep end. g_update computes the transposed tile (A/B swapped) so
// its per-tile f-broadcast is a plain per-lane load instead of 8 shuffles.
// ---------------------------------------------------------------------------

typedef float v2f __attribute__((ext_vector_type(2)));
typedef float v8f __attribute__((ext_vector_type(8)));

#define BB       8
#define SS       2048
#define PP       1024
#define EPS      0.01f
#define INV_EPS  100.0f
#define ITERS    50
#define NEG_INF  (-1.0e9f)

__device__ __forceinline__ void lse_combine(float& m, float& s, float m2, float s2) {
    float mn = fmaxf(m, m2);
    s = s * __expf(m - mn) + s2 * __expf(m2 - mn);
    m = mn;
}

// ------------------------------- init kernels ------------------------------

__global__ void zero_hg_kernel(float* __restrict__ hist, float* __restrict__ g) {
    int i = blockIdx.x * blockDim.x + threadIdx.x;
    if (i < BB * PP) { hist[i] = 0.0f; g[i] = 0.0f; }
}

__global__ void hist_kernel(const int* __restrict__ labels, float* __restrict__ hist) {
    int i = blockIdx.x * blockDim.x + threadIdx.x;
    if (i < BB * SS) {
        int b = i / SS;
        atomicAdd(&hist[b * PP + labels[i]], 1.0f);  // exact integer adds: deterministic
    }
}

__global__ void logb_kernel(const float* __restrict__ hist, float* __restrict__ logb) {
    int i = blockIdx.x * blockDim.x + threadIdx.x;
    if (i < BB * PP) {
        float h = hist[i];
        logb[i] = (h > 0.5f) ? __logf(h * (1.0f / (float)SS)) : NEG_INF;
    }
}

// ------------------------- f-update: LSE over P ----------------------------
// wave -> (b, s-tile of 16 rows); sweeps all P in 16-col WMMA tiles.
// Tile = C[s,p]: A rows from predictions, B cols from positions.
__global__ void f_update_kernel(const float* __restrict__ pred,
                                const float* __restrict__ pos,
                                const float* __restrict__ g,
                                float* __restrict__ f) {
    int wave = (blockIdx.x * blockDim.x + threadIdx.x) >> 5;
    int lane = threadIdx.x & 31;
    int b    = wave >> 7;          // SS/16 = 128 s-tiles per batch
    int s0   = (wave & 127) << 4;
    int m    = lane & 15;
    bool hi  = lane >= 16;

    // A fragment (16x4): row s0+m = [x0, x1, |x|^2, 1]
    float2 pxy = ((const float2*)pred)[b * SS + s0 + m];
    v2f a;
    if (!hi) { a.x = pxy.x;                         a.y = pxy.y; }
    else     { a.x = pxy.x*pxy.x + pxy.y*pxy.y;     a.y = 1.0f;  }

    float mArr[8], sArr[8];
#pragma unroll
    for (int i = 0; i < 8; ++i) { mArr[i] = -3.0e38f; sArr[i] = 0.0f; }

    // software pipeline: prefetch next tile's pos/g while crunching current
    float2 qxy = ((const float2*)pos)[b * PP + m];
    float  gv  = g[b * PP + m];

    for (int pt = 0; pt < PP / 16; ++pt) {
        float2 cq = qxy;
        float  cg = gv;
        if (pt + 1 < PP / 16) {  // uniform branch: EXEC stays all-ones
            int p1 = (pt + 1) << 4;
            qxy = ((const float2*)pos)[b * PP + p1 + m];
            gv  = g[b * PP + p1 + m];
        }
        // B fragment (4x16): col p0+m = [-2y0, -2y1, 1, |y|^2]
        v2f bf;
        if (!hi) { bf.x = -2.0f * cq.x;  bf.y = -2.0f * cq.y; }
        else     { bf.x = 1.0f;          bf.y = cq.x*cq.x + cq.y*cq.y; }

        v8f acc = {};
        acc = __builtin_amdgcn_wmma_f32_16x16x4_f32(false, a, false, bf,
                                                    (short)0, acc, false, false);
#pragma unroll
        for (int i = 0; i < 8; ++i) {
            float C  = fmaxf(acc[i], 0.0f);
            float x  = (cg - C) * INV_EPS;             // (mK + g/eps)
            float mn = fmaxf(mArr[i], x);
            sArr[i]  = sArr[i] * __expf(mArr[i] - mn) + __expf(x - mn);
            mArr[i]  = mn;
        }
    }
    // acc slot i: lanes 0-15 hold row i, lanes 16-31 row i+8 (cols = lane%16).
#pragma unroll
    for (int i = 0; i < 8; ++i) {
        float mm = mArr[i], ss = sArr[i];
#pragma unroll
        for (int k = 1; k < 16; k <<= 1) {
            float m2 = __shfl_xor(mm, k, 32);
            float s2 = __shfl_xor(ss, k, 32);
            lse_combine(mm, ss, m2, s2);
        }
        if (m == 0) {  // lane 0 -> row i, lane 16 -> row i+8
            float lse = mm + __logf(ss);
            int row   = i + (hi ? 8 : 0);
            f[b * SS + s0 + row] = EPS * (-__logf((float)SS) - lse);
        }
    }
}

// ------------------------- g-update: LSE over S ----------------------------
// wave -> (b, p-tile of 16 rows of C^T); sweeps all S in 16-col WMMA tiles.
// Transposed tile = C^T[p,s]: A rows from positions, B cols from predictions,
// so f[s0 + lane%16] is a plain per-lane load (no shuffles in the hot loop).
__global__ void g_update_kernel(const float* __restrict__ pred,
                                const float* __restrict__ pos,
                                const float* __restrict__ f,
                                const float* __restrict__ logb,
                                float* __restrict__ g) {
    int wave = (blockIdx.x * blockDim.x + threadIdx.x) >> 5;
    int lane = threadIdx.x & 31;
    int b    = wave >> 6;          // PP/16 = 64 p-tiles per batch
    int p0   = (wave & 63) << 4;
    int m    = lane & 15;
    bool hi  = lane >= 16;

    // A fragment (16x4): row p0+m = [y0, y1, |y|^2, 1]
    float2 qxy = ((const float2*)pos)[b * PP + p0 + m];
    v2f a;
    if (!hi) { a.x = qxy.x;                     a.y = qxy.y; }
    else     { a.x = qxy.x*qxy.x + qxy.y*qxy.y; a.y = 1.0f;  }

    float mArr[8], sArr[8];
#pragma unroll
    for (int i = 0; i < 8; ++i) { mArr[i] = -3.0e38f; sArr[i] = 0.0f; }

    float2 pxy = ((const float2*)pred)[b * SS + m];
    float  fv  = f[b * SS + m];

    for (int st = 0; st < SS / 16; ++st) {
        float2 cp = pxy;
        float  cf = fv;
        if (st + 1 < SS / 16) {  // uniform branch
            int s1 = (st + 1) << 4;
            pxy = ((const float2*)pred)[b * SS + s1 + m];
            fv  = f[b * SS + s1 + m];
        }
        // B fragment (4x16): col s0+m = [-2x0, -2x1, 1, |x|^2]
        v2f bf;
        if (!hi) { bf.x = -2.0f * cp.x;  bf.y = -2.0f * cp.y; }
        else     { bf.x = 1.0f;          bf.y = cp.x*cp.x + cp.y*cp.y; }

        v8f acc = {};
        acc = __builtin_amdgcn_wmma_f32_16x16x4_f32(false, a, false, bf,
                                                    (short)0, acc, false, false);
#pragma unroll
        for (int i = 0; i < 8; ++i) {
            float C  = fmaxf(acc[i], 0.0f);            // C^T[p,s] == C[s,p]
            float x  = (cf - C) * INV_EPS;             // (mK + f/eps)
            float mn = fmaxf(mArr[i], x);
            sArr[i]  = sArr[i] * __expf(mArr[i] - mn) + __expf(x - mn);
            mArr[i]  = mn;
        }
    }
    // acc slot i: lanes 0-15 hold p-row i, lanes 16-31 p-row i+8 (s = lane%16).
#pragma unroll
    for (int i = 0; i < 8; ++i) {
        float mm = mArr[i], ss = sArr[i];
#pragma unroll
        for (int k = 1; k < 16; k <<= 1) {
            float m2 = __shfl_xor(mm, k, 32);
            float s2 = __shfl_xor(ss, k, 32);
            lse_combine(mm, ss, m2, s2);
        }
        if (m == 0) {  // lane 0 -> p-row i, lane 16 -> p-row i+8
            float lse = mm + __logf(ss);
            int row   = i + (hi ? 8 : 0);
            g[b * PP + p0 + row] = EPS * (logb[b * PP + p0 + row] - lse);
        }
    }
}

// --------------------- final distance: sum exp(logT)*C ---------------------
// Runs once; per-tile shuffles for f-broadcast are fine here.
__global__ void dist_kernel(const float* __restrict__ pred,
                            const float* __restrict__ pos,
                            const float* __restrict__ f,
                            const float* __restrict__ g,
                            float* __restrict__ partials) {
    int wave = (blockIdx.x * blockDim.x + threadIdx.x) >> 5;
    int lane = threadIdx.x & 31;
    int b    = wave >> 7;
    int s0   = (wave & 127) << 4;
    int m    = lane & 15;
    bool hi  = lane >= 16;

    float2 pxy = ((const float2*)pred)[b * SS + s0 + m];
    v2f a;
    if (!hi) { a.x = pxy.x;                     a.y = pxy.y; }
    else     { a.x = pxy.x*pxy.x + pxy.y*pxy.y; a.y = 1.0f;  }

    float fv = f[b * SS + s0 + m];
    int srcBase = hi ? 8 : 0;
    float accsum = 0.0f;

    for (int pt = 0; pt < PP / 16; ++pt) {
        int p0 = pt << 4;
        float2 qxy = ((const float2*)pos)[b * PP + p0 + m];
        v2f bf;
        if (!hi) { bf.x = -2.0f * qxy.x;  bf.y = -2.0f * qxy.y; }
        else     { bf.x = 1.0f;           bf.y = qxy.x*qxy.x + qxy.y*qxy.y; }

        v8f acc = {};
        acc = __builtin_amdgcn_wmma_f32_16x16x4_f32(false, a, false, bf,
                                                    (short)0, acc, false, false);
        float gv = g[b * PP + p0 + m];
#pragma unroll
        for (int i = 0; i < 8; ++i) {
            float frow = __shfl(fv, srcBase + i, 32);  // f[s0 + i + 8*hi]
            float C    = fmaxf(acc[i], 0.0f);
            float t    = __expf((frow + gv - C) * INV_EPS);  // exp(logT)
            accsum += t * C;
        }
    }
    // wave sum (fixed order -> deterministic)
#pragma unroll
    for (int k = 1; k < 32; k <<= 1) accsum += __shfl_xor(accsum, k, 32);
    if (lane == 0) partials[wave] = accsum;
}

__global__ void reduce_kernel(const float* __restrict__ partials, float* __restrict__ out) {
    __shared__ float sm[256];
    float v = 0.0f;
    for (int i = threadIdx.x; i < BB * (SS / 16); i += 256) v += partials[i];
    sm[threadIdx.x] = v;
    __syncthreads();
    for (int s = 128; s > 0; s >>= 1) {
        if (threadIdx.x < s) sm[threadIdx.x] += sm[threadIdx.x + s];
        __syncthreads();
    }
    if (threadIdx.x == 0) out[0] = sm[0] * (1.0f / (float)BB);
}

// ------------------------------- launcher ----------------------------------

extern "C" void kernel_launch(void* const* d_in, const int* in_sizes, int n_in,
                              void* d_out, int out_size, void* d_ws, size_t ws_size,
                              hipStream_t stream) {
    const float* pred   = (const float*)d_in[0];   // [B,S,2]
    const int*   labels = (const int*)  d_in[1];   // [B,S]
    const float* pos    = (const float*)d_in[2];   // [B,P,2]
    float*       out    = (float*)d_out;

    // workspace layout (floats)
    char* ws = (char*)d_ws;
    float* f        = (float*)(ws);                          // B*S
    float* g        = (float*)(ws + BB * SS * 4);            // B*P
    float* logb     = (float*)(ws + BB * SS * 4 + BB * PP * 4);
    float* hist     = (float*)(ws + BB * SS * 4 + 2 * BB * PP * 4);
    float* partials = (float*)(ws + BB * SS * 4 + 3 * BB * PP * 4);  // B*S/16

    const int nWavesF = BB * (SS / 16);   // 1024
    const int nWavesG = BB * (PP / 16);   // 512

    zero_hg_kernel<<<(BB * PP + 255) / 256, 256, 0, stream>>>(hist, g);
    hist_kernel  <<<(BB * SS + 255) / 256, 256, 0, stream>>>(labels, hist);
    logb_kernel  <<<(BB * PP + 255) / 256, 256, 0, stream>>>(hist, logb);

    for (int it = 0; it < ITERS; ++it) {
        f_update_kernel<<<nWavesF * 32 / 256, 256, 0, stream>>>(pred, pos, g, f);
        g_update_kernel<<<nWavesG * 32 / 256, 256, 0, stream>>>(pred, pos, f, logb, g);
    }

    dist_kernel  <<<nWavesF * 32 / 256, 256, 0, stream>>>(pred, pos, f, g, partials);
    reduce_kernel<<<1, 256, 0, stream>>>(partials, out);
}